// LinearAttention_63333587747477
// MI455X (gfx1250) — compile-verified
//
#include <hip/hip_runtime.h>
#include <hip/hip_bf16.h>

typedef __attribute__((ext_vector_type(2))) float v2f;
typedef __attribute__((ext_vector_type(8))) float v8f;
typedef __attribute__((ext_vector_type(4))) unsigned int v4u;
typedef __attribute__((ext_vector_type(8))) int v8i;
typedef __attribute__((ext_vector_type(4))) int v4i;

// D(16x16,f32) = A(16x4,f32) x B(4x16,f32) + C  — native fp32 matrix op on CDNA5
static __device__ __forceinline__ v8f wmma4(v2f a, v2f b, v8f c) {
    return __builtin_amdgcn_wmma_f32_16x16x4_f32(
        /*neg_a=*/false, a, /*neg_b=*/false, b,
        /*c_mod=*/(short)0, c, /*reuse_a=*/false, /*reuse_b=*/false);
}

// TDM: DMA a 2D tile (rows x width f32, row stride = stride_elts) from global
// memory into LDS at byte offset lds_off. Tracked by TENSORcnt.
static __device__ __forceinline__ void tdm_load_2d_f32(
    unsigned int lds_off, const float* gaddr,
    unsigned int width_elts, unsigned int rows,
    unsigned int tensor_rows, unsigned int stride_elts)
{
    const unsigned long long ga = (unsigned long long)(uintptr_t)gaddr;
    v4u g0;
    g0[0] = 1u;                                         // count=1, normal mode
    g0[1] = lds_off;                                    // lds_addr (bytes)
    g0[2] = (unsigned int)(ga & 0xffffffffu);           // global_addr[31:0]
    g0[3] = (unsigned int)((ga >> 32) & 0x01ffffffu)    // global_addr[56:32]
          | (2u << 30);                                 // type = 2 ("image")
    v8i g1;
    g1[0] = (int)(2u << 16);                            // data_size = 4 bytes
    g1[1] = (int)((width_elts & 0xffffu) << 16);        // tensor_dim0[15:0]
    g1[2] = (int)((width_elts >> 16)
          | ((tensor_rows & 0xffffu) << 16));           // dim0[31:16] | dim1[15:0]
    g1[3] = (int)((tensor_rows >> 16)
          | ((width_elts & 0xffffu) << 16));            // dim1[31:16] | tile_dim0
    g1[4] = (int)(rows & 0xffffu);                      // tile_dim1 (tile_dim2=0)
    g1[5] = (int)stride_elts;                           // tensor_dim0_stride[31:0]
    g1[6] = 0;
    g1[7] = 0;
    const v4i z4 = {0, 0, 0, 0};
#if defined(__clang_major__) && (__clang_major__ >= 23)
    const v8i z8 = {0, 0, 0, 0, 0, 0, 0, 0};
    __builtin_amdgcn_tensor_load_to_lds(g0, g1, z4, z4, z8, 0);
#else
    __builtin_amdgcn_tensor_load_to_lds(g0, g1, z4, z4, 0);
#endif
}

// Shapes (fixed by the reference)
#define BATCH   4
#define SEQ     8192
#define DIM     128
#define HEADS   8
#define DH      64
#define INNER   512            // HEADS*DH
#define NCOLS   1536           // 3*INNER
#define ROWS    32768          // BATCH*SEQ

// ---------------------------------------------------------------------------
// Kernel 1: qkv = x @ w_qkv fused with InstanceNorm(k), InstanceNorm(v).
// One block = 16 rows x 1536 cols. 8 waves, each wave owns 12 N-tiles
// (192 contiguous cols = exactly 3 head-chunks of 64, never splitting one).
// LDS: 16*1536 f32 = 96KB (dynamic).
// ---------------------------------------------------------------------------
__global__ __launch_bounds__(256) void qkv_norm_kernel(
    const float* __restrict__ x, const float* __restrict__ wqkv,
    float* __restrict__ q_ws, float* __restrict__ k_ws, float* __restrict__ v_ws)
{
    extern __shared__ float smem[];              // [16][1536]
    const int r0   = blockIdx.x * 16;            // global row block
    const int bidx = r0 >> 13;                   // batch index
    const int nl   = r0 & (SEQ - 1);             // n offset within batch
    const int lane = threadIdx.x & 31;
    const int wv   = threadIdx.x >> 5;           // wave 0..7
    const int col  = lane & 15;
    const int half = lane >> 4;

    v8f acc[12] = {};
    const float* xa = x + (size_t)r0 * DIM;

    #pragma unroll 1
    for (int k0 = 0; k0 < DIM; k0 += 4) {
        v2f a;
        a.x = xa[(size_t)col * DIM + k0 + 2 * half];        // A row = lane&15
        a.y = xa[(size_t)col * DIM + k0 + 1 + 2 * half];
        #pragma unroll
        for (int t = 0; t < 12; ++t) {
            const int c0 = (wv * 12 + t) * 16;
            v2f bb;
            bb.x = wqkv[(size_t)(k0 + 2 * half) * NCOLS + c0 + col];
            bb.y = wqkv[(size_t)(k0 + 1 + 2 * half) * NCOLS + c0 + col];
            acc[t] = wmma4(a, bb, acc[t]);
        }
    }

    // accumulators -> LDS tile
    #pragma unroll
    for (int t = 0; t < 12; ++t) {
        const int c0 = (wv * 12 + t) * 16;
        #pragma unroll
        for (int i = 0; i < 8; ++i) {
            const int rr = i + 8 * half;
            smem[rr * NCOLS + c0 + col] = acc[t][i];
        }
    }
    __syncthreads();

    // q passthrough: cols [0,512) -> (b,h,n,d) layout
    #pragma unroll 4
    for (int i = 0; i < 32; ++i) {
        const int idx = threadIdx.x + 256 * i;   // 0..8191 over 16x512
        const int rr = idx >> 9;
        const int cc = idx & 511;
        const int h  = cc >> 6;
        const int d  = cc & 63;
        q_ws[((size_t)(bidx * HEADS + h) * SEQ + nl + rr) * DH + d] =
            smem[rr * NCOLS + cc];
    }

    // InstanceNorm over each 64-chunk of k (cols 512..1023) and v (1024..1535).
    // 256 threads <-> 16 rows x 16 chunks. Skewed LDS reads avoid bank clashes.
    {
        const int rr    = threadIdx.x >> 4;
        const int chunk = threadIdx.x & 15;
        const int isv   = chunk >> 3;
        const int h     = chunk & 7;
        const int base  = INNER + chunk * 64;
        float s = 0.f, s2 = 0.f;
        #pragma unroll 8
        for (int j = 0; j < 64; ++j) {
            const float vv = smem[rr * NCOLS + base + ((j + chunk) & 63)];
            s += vv; s2 += vv * vv;
        }
        const float mu   = s * (1.0f / 64.0f);
        const float var  = s2 * (1.0f / 64.0f) - mu * mu;    // biased variance
        const float rinv = rsqrtf(var + 1e-5f);
        float* dst = (isv ? v_ws : k_ws) +
                     ((size_t)(bidx * HEADS + h) * SEQ + nl + rr) * DH;
        #pragma unroll 8
        for (int j = 0; j < 64; ++j) {
            const int d = (j + chunk) & 63;
            dst[d] = (smem[rr * NCOLS + base + d] - mu) * rinv;
        }
    }
}

// ---------------------------------------------------------------------------
// Kernel 2: dots[bh] += k[bh]^T @ v[bh]  (64x64 per bh, K reduced over n).
// Grid = 32 bh x 16 n-chunks (512 n each). k/v tiles (64 rows x 64 f32 each)
// are DMA'd into LDS by the Tensor Data Mover (wave 0 issues, TENSORcnt
// waited, block barrier publishes), then 8 waves consume WMMA fragments from
// LDS. Partials merged with global f32 atomics into pre-zeroed dots buffer.
// ---------------------------------------------------------------------------
__global__ __launch_bounds__(256) void ktv_kernel(
    const float* __restrict__ k_ws, const float* __restrict__ v_ws,
    float* __restrict__ dots)
{
    __shared__ float klds[64 * 64];              // 16KB
    __shared__ float vlds[64 * 64];              // 16KB
    const int bh    = blockIdx.x >> 4;
    const int chunk = blockIdx.x & 15;
    const int lane  = threadIdx.x & 31;
    const int wv    = threadIdx.x >> 5;
    const int col   = lane & 15;
    const int half  = lane >> 4;
    const int d0    = (wv & 3) * 16;
    const int e0b   = (wv >> 2) * 32;
    const size_t base = (size_t)bh * SEQ * DH;
    const int n0 = chunk * 512;

    // LDS byte offsets (low 32 bits of the flat address, ISA 10.2)
    const unsigned int koff = (unsigned int)(uintptr_t)&klds[0];
    const unsigned int voff = (unsigned int)(uintptr_t)&vlds[0];

    v8f acc[2] = {};
    #pragma unroll 1
    for (int tile = 0; tile < 8; ++tile) {
        const int nbase = n0 + tile * 64;
        if (wv == 0) {                           // one wave issues the DMA
            tdm_load_2d_f32(koff, k_ws + base + (size_t)nbase * DH,
                            DH, 64, 64, DH);
            tdm_load_2d_f32(voff, v_ws + base + (size_t)nbase * DH,
                            DH, 64, 64, DH);
            __builtin_amdgcn_s_wait_tensorcnt(0);
        }
        __syncthreads();                         // tiles visible to all waves

        #pragma unroll 4
        for (int nn = 0; nn < 64; nn += 4) {
            v2f a;                               // A = k^T tile (d x n)
            a.x = klds[(nn + 2 * half) * DH + d0 + col];
            a.y = klds[(nn + 1 + 2 * half) * DH + d0 + col];
            #pragma unroll
            for (int t = 0; t < 2; ++t) {
                const int e0 = e0b + t * 16;
                v2f bb;                          // B = v tile (n x e)
                bb.x = vlds[(nn + 2 * half) * DH + e0 + col];
                bb.y = vlds[(nn + 1 + 2 * half) * DH + e0 + col];
                acc[t] = wmma4(a, bb, acc[t]);
            }
        }
        __syncthreads();                         // done reading before next DMA
    }

    #pragma unroll
    for (int t = 0; t < 2; ++t) {
        const int e0 = e0b + t * 16;
        #pragma unroll
        for (int i = 0; i < 8; ++i) {
            const int rr = i + 8 * half;
            unsafeAtomicAdd(&dots[(size_t)bh * 4096 + (d0 + rr) * 64 + e0 + col],
                            acc[t][i]);
        }
    }
}

// ---------------------------------------------------------------------------
// Kernel 3: out = concat_h(q @ dots_h / n) @ w_out + b_out, fused.
// Stage 1: wave h computes T_h(16x64) = Q(16x64) @ dots_h, scaled, -> LDS.
// Stage 2: wave w computes out tile (16x16) = T(16x512) @ w_out + bias,
//          bias preloaded into the WMMA C accumulator.
// LDS pitch 516 (516 % 64 == 4) -> conflict-free transposed stage-2 reads.
// ---------------------------------------------------------------------------
#define TPITCH 516
__global__ __launch_bounds__(256) void out_kernel(
    const float* __restrict__ q_ws, const float* __restrict__ dots,
    const float* __restrict__ w_out, const float* __restrict__ b_out,
    float* __restrict__ out)
{
    extern __shared__ float smem[];              // [16][TPITCH]
    const int r0   = blockIdx.x * 16;
    const int bidx = r0 >> 13;
    const int nl   = r0 & (SEQ - 1);
    const int lane = threadIdx.x & 31;
    const int wv   = threadIdx.x >> 5;
    const int col  = lane & 15;
    const int half = lane >> 4;

    // ---- stage 1: per-head q @ dots ----
    {
        const int h  = wv;
        const int bh = bidx * HEADS + h;
        const size_t qb = (size_t)bh * SEQ * DH;
        const float* dd = dots + (size_t)bh * 4096;
        v8f acc[4] = {};
        #pragma unroll 1
        for (int k0 = 0; k0 < DH; k0 += 4) {
            v2f a;
            a.x = q_ws[qb + (size_t)(nl + col) * DH + k0 + 2 * half];
            a.y = q_ws[qb + (size_t)(nl + col) * DH + k0 + 1 + 2 * half];
            #pragma unroll
            for (int t = 0; t < 4; ++t) {
                v2f bb;
                bb.x = dd[(k0 + 2 * half) * 64 + t * 16 + col];
                bb.y = dd[(k0 + 1 + 2 * half) * 64 + t * 16 + col];
                acc[t] = wmma4(a, bb, acc[t]);
            }
        }
        const float scale = 1.0f / (float)SEQ;
        #pragma unroll
        for (int t = 0; t < 4; ++t) {
            #pragma unroll
            for (int i = 0; i < 8; ++i) {
                const int rr = i + 8 * half;
                smem[rr * TPITCH + h * DH + t * 16 + col] = acc[t][i] * scale;
            }
        }
    }
    __syncthreads();

    // ---- stage 2: T @ w_out + b_out ----
    {
        const int n0 = wv * 16;
        const float bias = b_out[n0 + col];
        v8f acc;
        #pragma unroll
        for (int i = 0; i < 8; ++i) acc[i] = bias;   // bias rides in C
        #pragma unroll 2
        for (int k0 = 0; k0 < INNER; k0 += 4) {
            v2f a;                                    // A row = lane&15
            a.x = smem[col * TPITCH + k0 + 2 * half];
            a.y = smem[col * TPITCH + k0 + 1 + 2 * half];
            v2f bb;
            bb.x = w_out[(size_t)(k0 + 2 * half) * DIM + n0 + col];
            bb.y = w_out[(size_t)(k0 + 1 + 2 * half) * DIM + n0 + col];
            acc = wmma4(a, bb, acc);
        }
        #pragma unroll
        for (int i = 0; i < 8; ++i) {
            const int rr = i + 8 * half;
            out[(size_t)(r0 + rr) * DIM + n0 + col] = acc[i];
        }
    }
}

// ---------------------------------------------------------------------------
// Host launcher. Workspace layout (floats): q | k | v | dots  (~192.5 MB)
// ---------------------------------------------------------------------------
extern "C" void kernel_launch(void* const* d_in, const int* in_sizes, int n_in,
                              void* d_out, int out_size, void* d_ws, size_t ws_size,
                              hipStream_t stream) {
    const float* x     = (const float*)d_in[0];
    const float* w_qkv = (const float*)d_in[1];
    const float* w_out = (const float*)d_in[2];
    const float* b_out = (const float*)d_in[3];
    float* out = (float*)d_out;

    const size_t per = (size_t)BATCH * HEADS * SEQ * DH;   // 16,777,216 floats
    float* q_ws  = (float*)d_ws;
    float* k_ws  = q_ws + per;
    float* v_ws  = k_ws + per;
    float* dots  = v_ws + per;                              // 32*64*64 floats

    hipMemsetAsync(dots, 0, (size_t)BATCH * HEADS * 64 * 64 * sizeof(float), stream);

    qkv_norm_kernel<<<ROWS / 16, 256, 16 * NCOLS * sizeof(float), stream>>>(
        x, w_qkv, q_ws, k_ws, v_ws);

    ktv_kernel<<<BATCH * HEADS * 16, 256, 0, stream>>>(k_ws, v_ws, dots);

    out_kernel<<<ROWS / 16, 256, 16 * TPITCH * sizeof(float), stream>>>(
        q_ws, dots, w_out, b_out, out);
}